// get_model_55078660604125
// MI455X (gfx1250) — compile-verified
//
#include <hip/hip_runtime.h>

typedef float v2f __attribute__((ext_vector_type(2)));
typedef float v8f __attribute__((ext_vector_type(8)));

#define B_    8
#define N_    8192
#define D_    6
#define S_    1024
#define NS_   32
#define R2_   0.04f
#define NCOL  (B_ * S_ * NS_)          /* 262144 columns */
#define EPS_  1e-5f
#define CNTF  ((float)NCOL)

/* ---------------- workspace layout (bytes) ---------------- */
#define OFF_FPS   ((size_t)0)                          /* B*S ints   = 32 KB  */
#define OFF_STATS ((size_t)32768)                      /* 512 floats = 2 KB   */
#define OFF_X0    ((size_t)(32768 + 2048))             /* 9  * NCOL floats    */
#define OFF_Y0    (OFF_X0 + (size_t)9  * NCOL * 4)     /* 64 * NCOL floats    */
#define OFF_Y1    (OFF_Y0 + (size_t)64 * NCOL * 4)     /* 64 * NCOL floats    */

/* ds_swizzle group-of-32 xor-butterfly: offset = xor<<10 | or<<5 | and(0x1f) */
#define SWZ(x, m) __builtin_amdgcn_ds_swizzle((x), ((m) << 10) | 0x1f)

/* butterfly-sum within each 16-lane half of the wave (WMMA D-row group) */
__device__ __forceinline__ float red16(float v) {
  v += __int_as_float(SWZ(__float_as_int(v), 1));
  v += __int_as_float(SWZ(__float_as_int(v), 2));
  v += __int_as_float(SWZ(__float_as_int(v), 4));
  v += __int_as_float(SWZ(__float_as_int(v), 8));
  return v;
}

/* butterfly-max within each 16-lane half of the wave */
__device__ __forceinline__ float rmax16(float v) {
  v = fmaxf(v, __int_as_float(SWZ(__float_as_int(v), 1)));
  v = fmaxf(v, __int_as_float(SWZ(__float_as_int(v), 2)));
  v = fmaxf(v, __int_as_float(SWZ(__float_as_int(v), 4)));
  v = fmaxf(v, __int_as_float(SWZ(__float_as_int(v), 8)));
  return v;
}

/* ================= 0: zero the per-layer BN statistics ================= */
__global__ void zero_stats_kernel(float* stats) {
  int t = blockIdx.x * blockDim.x + threadIdx.x;
  if (t < 512) stats[t] = 0.f;
}

/* ================= 1: farthest point sampling ================= */
/* one 1024-thread block per batch; each thread owns 8 points in registers */
__global__ __launch_bounds__(1024) void fps_kernel(const float* __restrict__ xyz,
                                                   int* __restrict__ fpsIdx,
                                                   float* __restrict__ newXyz) {
  const int b = blockIdx.x, t = threadIdx.x;
  const float* xb = xyz + (size_t)b * 3 * N_;
  float px[8], py[8], pz[8], pd[8];
#pragma unroll
  for (int j = 0; j < 8; ++j) {
    int n = t + j * 1024;
    px[j] = xb[n]; py[j] = xb[N_ + n]; pz[j] = xb[2 * N_ + n];
    pd[j] = 1e10f;
  }
  __shared__ float sv[1024];
  __shared__ int   si[1024];
  int f = 0;
  for (int i = 0; i < S_; ++i) {
    if (t == 0) {
      fpsIdx[b * S_ + i] = f;
      newXyz[(size_t)b * 3 * S_ + 0 * S_ + i] = xb[f];
      newXyz[(size_t)b * 3 * S_ + 1 * S_ + i] = xb[N_ + f];
      newXyz[(size_t)b * 3 * S_ + 2 * S_ + i] = xb[2 * N_ + f];
    }
    float cx = xb[f], cy = xb[N_ + f], cz = xb[2 * N_ + f];
    float bv = -1.f; int bi = 0;
#pragma unroll
    for (int j = 0; j < 8; ++j) {
      float dx = px[j] - cx, dy = py[j] - cy, dz = pz[j] - cz;
      float d = dx * dx + dy * dy + dz * dz;
      pd[j] = fminf(pd[j], d);
      if (pd[j] > bv) { bv = pd[j]; bi = t + j * 1024; }
    }
    sv[t] = bv; si[t] = bi;
    __syncthreads();
    for (int s = 512; s > 0; s >>= 1) {
      if (t < s) {
        float ov = sv[t + s]; int oi = si[t + s];
        if (ov > sv[t] || (ov == sv[t] && oi < si[t])) { sv[t] = ov; si[t] = oi; }
      }
      __syncthreads();
    }
    f = si[0];
    __syncthreads();
  }
}

/* ================= 2: ball query + grouping -> X0 [9][NCOL] ================= */
/* one wave per (b,s) group; ballot-compaction picks first 32 in-radius indices */
__global__ __launch_bounds__(256) void group_kernel(const float* __restrict__ xyz,
                                                    const float* __restrict__ pts,
                                                    const int* __restrict__ fpsIdx,
                                                    float* __restrict__ X0) {
  const int lane = threadIdx.x & 31;
  const int g = blockIdx.x * 8 + (threadIdx.x >> 5);   /* group = b*S + s */
  const int b = g >> 10;
  const float* xb = xyz + (size_t)b * 3 * N_;
  const int cidx = fpsIdx[g];
  const float cx = xb[cidx], cy = xb[N_ + cidx], cz = xb[2 * N_ + cidx];

  int cnt = 0, myIdx = -1, firstIdx = -1;
  for (int j0 = 0; j0 < N_ && cnt < NS_; j0 += 32) {
    int j = j0 + lane;
    float dx = xb[j] - cx, dy = xb[N_ + j] - cy, dz = xb[2 * N_ + j] - cz;
    bool within = (dx * dx + dy * dy + dz * dz) <= R2_;
    unsigned m = (unsigned)__ballot(within);
    if (m) {
      if (firstIdx < 0) firstIdx = j0 + __ffs(m) - 1;
      unsigned mm = m;
      while (mm && cnt < NS_) {            /* pop bits in ascending index order */
        int bit = __ffs(mm) - 1; mm &= mm - 1;
        if (cnt == lane) myIdx = j0 + bit; /* slot number == cnt, uniform */
        ++cnt;
      }
    }
  }
  if (myIdx < 0) myIdx = (firstIdx >= 0) ? firstIdx : cidx;  /* pad like ref */

  const size_t col = (size_t)g * NS_ + lane;
  X0[0 * (size_t)NCOL + col] = xb[myIdx] - cx;
  X0[1 * (size_t)NCOL + col] = xb[N_ + myIdx] - cy;
  X0[2 * (size_t)NCOL + col] = xb[2 * N_ + myIdx] - cz;
  const float* pb = pts + (size_t)b * D_ * N_;
#pragma unroll
  for (int d = 0; d < D_; ++d)
    X0[(size_t)(3 + d) * NCOL + col] = pb[(size_t)d * N_ + myIdx];
}

/* =============== WMMA f32 16x16x4 fragment helpers (ISA 7.12.2) =============== */
__device__ __forceinline__ v8f wmma_f32(v2f a, v2f b, v8f c) {
  return __builtin_amdgcn_wmma_f32_16x16x4_f32(false, a, false, b, (short)0, c,
                                               false, false);
}

/* ================= 3: layer0 GEMM  Y0 = W0[64x9] @ X0 + b0, stats0 ========== */
__global__ __launch_bounds__(256) void gemm0_kernel(const float* __restrict__ X0,
                                                    const float* __restrict__ w0,
                                                    const float* __restrict__ b0,
                                                    float* __restrict__ Y0,
                                                    float* __restrict__ gsum,
                                                    float* __restrict__ gsqs) {
  __shared__ float Wl[64 * 13];      /* K padded 9->12, row stride 13 */
  __shared__ float Xl[12 * 64];
  __shared__ float sSum[64], sSqs[64];
  const int tid = threadIdx.x;
  for (int i = tid; i < 64 * 13; i += 256) {
    int m = i / 13, k = i % 13;
    Wl[i] = (k < 9) ? w0[m * 9 + k] : 0.f;
  }
  if (tid < 64) { sSum[tid] = 0.f; sSqs[tid] = 0.f; }
  const size_t colBase = (size_t)blockIdx.x * 64;
  for (int i = tid; i < 12 * 64; i += 256) {
    int r = i / 64, c = i % 64;
    Xl[i] = (r < 9) ? X0[(size_t)r * NCOL + colBase + c] : 0.f;
  }
  __syncthreads();

  const int wv = tid >> 5, lane = tid & 31, half = lane >> 4, l = lane & 15;
  const int t0 = wv * 2;
  const int m0 = (t0 >> 2) * 16, n0 = (t0 & 3) * 16;
  const int m1 = ((t0 + 1) >> 2) * 16, n1 = ((t0 + 1) & 3) * 16;
  v8f acc0 = {}, acc1 = {};
#pragma unroll
  for (int k0 = 0; k0 < 12; k0 += 4) {
    int kk = k0 + 2 * half;
    v2f a0 = { Wl[(m0 + l) * 13 + kk], Wl[(m0 + l) * 13 + kk + 1] };
    v2f bb0 = { Xl[kk * 64 + n0 + l], Xl[(kk + 1) * 64 + n0 + l] };
    acc0 = wmma_f32(a0, bb0, acc0);
    v2f a1 = { Wl[(m1 + l) * 13 + kk], Wl[(m1 + l) * 13 + kk + 1] };
    v2f bb1 = { Xl[kk * 64 + n1 + l], Xl[(kk + 1) * 64 + n1 + l] };
    acc1 = wmma_f32(a1, bb1, acc1);
  }
#pragma unroll
  for (int v = 0; v < 8; ++v) {
    int ch = m0 + v + 8 * half, c = n0 + l;
    float val = acc0[v] + b0[ch];
    Y0[(size_t)ch * NCOL + colBase + c] = val;
    float s = red16(val), q = red16(val * val);
    if (l == 0) { atomicAdd(&sSum[ch], s); atomicAdd(&sSqs[ch], q); }
    ch = m1 + v + 8 * half; c = n1 + l;
    val = acc1[v] + b0[ch];
    Y0[(size_t)ch * NCOL + colBase + c] = val;
    s = red16(val); q = red16(val * val);
    if (l == 0) { atomicAdd(&sSum[ch], s); atomicAdd(&sSqs[ch], q); }
  }
  __syncthreads();
  if (tid < 64) { atomicAdd(&gsum[tid], sSum[tid]); atomicAdd(&gsqs[tid], sSqs[tid]); }
}

/* ======= 4: layer1 GEMM  Y1 = W1[64x64] @ BN0ReLU(Y0) + b1, stats1 ======= */
__global__ __launch_bounds__(256) void gemm1_kernel(const float* __restrict__ Y0,
                                                    const float* __restrict__ w1,
                                                    const float* __restrict__ b1,
                                                    const float* __restrict__ g0,
                                                    const float* __restrict__ bt0,
                                                    const float* __restrict__ sum0,
                                                    const float* __restrict__ sqs0,
                                                    float* __restrict__ Y1,
                                                    float* __restrict__ gsum,
                                                    float* __restrict__ gsqs) {
  __shared__ float Wl[64 * 65];
  __shared__ float Xl[64 * 64];
  __shared__ float sSum[64], sSqs[64], sSc[64], sSh[64];
  const int tid = threadIdx.x;
  for (int i = tid; i < 64 * 65; i += 256) {
    int m = i / 65, k = i % 65;
    Wl[i] = (k < 64) ? w1[m * 64 + k] : 0.f;
  }
  if (tid < 64) {
    float mean = sum0[tid] / CNTF;
    float var = sqs0[tid] / CNTF - mean * mean;
    float sc = g0[tid] * rsqrtf(var + EPS_);
    sSc[tid] = sc; sSh[tid] = bt0[tid] - mean * sc;
    sSum[tid] = 0.f; sSqs[tid] = 0.f;
  }
  __syncthreads();
  const size_t colBase = (size_t)blockIdx.x * 64;
  for (int i = tid; i < 64 * 64; i += 256) {
    int r = i / 64, c = i % 64;
    Xl[i] = fmaxf(Y0[(size_t)r * NCOL + colBase + c] * sSc[r] + sSh[r], 0.f);
  }
  __syncthreads();

  const int wv = tid >> 5, lane = tid & 31, half = lane >> 4, l = lane & 15;
  const int t0 = wv * 2;
  const int m0 = (t0 >> 2) * 16, n0 = (t0 & 3) * 16;
  const int m1 = ((t0 + 1) >> 2) * 16, n1 = ((t0 + 1) & 3) * 16;
  v8f acc0 = {}, acc1 = {};
#pragma unroll
  for (int k0 = 0; k0 < 64; k0 += 4) {
    int kk = k0 + 2 * half;
    v2f a0 = { Wl[(m0 + l) * 65 + kk], Wl[(m0 + l) * 65 + kk + 1] };
    v2f bb0 = { Xl[kk * 64 + n0 + l], Xl[(kk + 1) * 64 + n0 + l] };
    acc0 = wmma_f32(a0, bb0, acc0);
    v2f a1 = { Wl[(m1 + l) * 65 + kk], Wl[(m1 + l) * 65 + kk + 1] };
    v2f bb1 = { Xl[kk * 64 + n1 + l], Xl[(kk + 1) * 64 + n1 + l] };
    acc1 = wmma_f32(a1, bb1, acc1);
  }
#pragma unroll
  for (int v = 0; v < 8; ++v) {
    int ch = m0 + v + 8 * half, c = n0 + l;
    float val = acc0[v] + b1[ch];
    Y1[(size_t)ch * NCOL + colBase + c] = val;
    float s = red16(val), q = red16(val * val);
    if (l == 0) { atomicAdd(&sSum[ch], s); atomicAdd(&sSqs[ch], q); }
    ch = m1 + v + 8 * half; c = n1 + l;
    val = acc1[v] + b1[ch];
    Y1[(size_t)ch * NCOL + colBase + c] = val;
    s = red16(val); q = red16(val * val);
    if (l == 0) { atomicAdd(&sSum[ch], s); atomicAdd(&sSqs[ch], q); }
  }
  __syncthreads();
  if (tid < 64) { atomicAdd(&gsum[tid], sSum[tid]); atomicAdd(&gsqs[tid], sSqs[tid]); }
}

/* ======= 5: layer2 stats pass — Y2 = W2[128x64] @ BN1ReLU(Y1) + b2 ======= */
__global__ __launch_bounds__(256) void gemm2_stats_kernel(const float* __restrict__ Y1,
                                                          const float* __restrict__ w2,
                                                          const float* __restrict__ b2,
                                                          const float* __restrict__ g1,
                                                          const float* __restrict__ bt1,
                                                          const float* __restrict__ sum1,
                                                          const float* __restrict__ sqs1,
                                                          float* __restrict__ gsum,
                                                          float* __restrict__ gsqs) {
  __shared__ float Wl[128 * 65];
  __shared__ float Xl[64 * 64];
  __shared__ float sSum[128], sSqs[128], sSc[64], sSh[64];
  const int tid = threadIdx.x;
  for (int i = tid; i < 128 * 65; i += 256) {
    int m = i / 65, k = i % 65;
    Wl[i] = (k < 64) ? w2[m * 64 + k] : 0.f;
  }
  if (tid < 64) {
    float mean = sum1[tid] / CNTF;
    float var = sqs1[tid] / CNTF - mean * mean;
    float sc = g1[tid] * rsqrtf(var + EPS_);
    sSc[tid] = sc; sSh[tid] = bt1[tid] - mean * sc;
  }
  if (tid < 128) { sSum[tid] = 0.f; sSqs[tid] = 0.f; }
  __syncthreads();
  const size_t colBase = (size_t)blockIdx.x * 64;
  for (int i = tid; i < 64 * 64; i += 256) {
    int r = i / 64, c = i % 64;
    Xl[i] = fmaxf(Y1[(size_t)r * NCOL + colBase + c] * sSc[r] + sSh[r], 0.f);
  }
  __syncthreads();

  const int wv = tid >> 5, lane = tid & 31, half = lane >> 4, l = lane & 15;
  const int mB = wv * 16;                 /* each wave owns one 16-row band */
  v8f acc[4] = {v8f{}, v8f{}, v8f{}, v8f{}};
#pragma unroll
  for (int k0 = 0; k0 < 64; k0 += 4) {
    int kk = k0 + 2 * half;
    v2f a = { Wl[(mB + l) * 65 + kk], Wl[(mB + l) * 65 + kk + 1] };
#pragma unroll
    for (int nt = 0; nt < 4; ++nt) {
      v2f bb = { Xl[kk * 64 + nt * 16 + l], Xl[(kk + 1) * 64 + nt * 16 + l] };
      acc[nt] = wmma_f32(a, bb, acc[nt]);
    }
  }
#pragma unroll
  for (int v = 0; v < 8; ++v) {
    int ch = mB + v + 8 * half;
    /* sum the 4 N-tiles per lane first, then one cross-lane reduction */
    float vs = 0.f, vq = 0.f;
#pragma unroll
    for (int nt = 0; nt < 4; ++nt) {
      float val = acc[nt][v] + b2[ch];
      vs += val; vq += val * val;
    }
    float s = red16(vs), q = red16(vq);
    if (l == 0) { atomicAdd(&sSum[ch], s); atomicAdd(&sSqs[ch], q); }
  }
  __syncthreads();
  if (tid < 128) { atomicAdd(&gsum[tid], sSum[tid]); atomicAdd(&gsqs[tid], sSqs[tid]); }
}

/* ===== 6: recompute Y2 per (b,s), BN2+ReLU, max over ns=32 -> feat ===== */
__global__ __launch_bounds__(256) void final_kernel(const float* __restrict__ Y1,
                                                    const float* __restrict__ w2,
                                                    const float* __restrict__ b2,
                                                    const float* __restrict__ g1,
                                                    const float* __restrict__ bt1,
                                                    const float* __restrict__ sum1,
                                                    const float* __restrict__ sqs1,
                                                    const float* __restrict__ g2,
                                                    const float* __restrict__ bt2,
                                                    const float* __restrict__ sum2,
                                                    const float* __restrict__ sqs2,
                                                    float* __restrict__ feat) {
  __shared__ float Wl[128 * 65];
  __shared__ float Xl[64 * 32];
  __shared__ float sSc1[64], sSh1[64], sSc2[128], sSh2[128];
  const int tid = threadIdx.x;
  for (int i = tid; i < 128 * 65; i += 256) {
    int m = i / 65, k = i % 65;
    Wl[i] = (k < 64) ? w2[m * 64 + k] : 0.f;
  }
  if (tid < 64) {
    float mean = sum1[tid] / CNTF;
    float var = sqs1[tid] / CNTF - mean * mean;
    float sc = g1[tid] * rsqrtf(var + EPS_);
    sSc1[tid] = sc; sSh1[tid] = bt1[tid] - mean * sc;
  }
  if (tid < 128) {
    float mean = sum2[tid] / CNTF;
    float var = sqs2[tid] / CNTF - mean * mean;
    float sc = g2[tid] * rsqrtf(var + EPS_);
    sSc2[tid] = sc; sSh2[tid] = bt2[tid] - mean * sc;
  }
  __syncthreads();
  const int g = blockIdx.x;                 /* group = b*S + s */
  const size_t colBase = (size_t)g * NS_;
  for (int i = tid; i < 64 * 32; i += 256) {
    int r = i / 32, c = i % 32;
    Xl[i] = fmaxf(Y1[(size_t)r * NCOL + colBase + c] * sSc1[r] + sSh1[r], 0.f);
  }
  __syncthreads();

  const int wv = tid >> 5, lane = tid & 31, half = lane >> 4, l = lane & 15;
  const int mB = wv * 16;
  v8f acc0 = {}, acc1 = {};
#pragma unroll
  for (int k0 = 0; k0 < 64; k0 += 4) {
    int kk = k0 + 2 * half;
    v2f a = { Wl[(mB + l) * 65 + kk], Wl[(mB + l) * 65 + kk + 1] };
    v2f bb0 = { Xl[kk * 32 + 0 + l], Xl[(kk + 1) * 32 + 0 + l] };
    acc0 = wmma_f32(a, bb0, acc0);
    v2f bb1 = { Xl[kk * 32 + 16 + l], Xl[(kk + 1) * 32 + 16 + l] };
    acc1 = wmma_f32(a, bb1, acc1);
  }
  /* BN2 + ReLU, then max over the 32 columns of this (b,s) group.
     Per VGPR v the 32 samples live in: acc0 (cols 0-15) and acc1 (cols 16-31),
     striped across the 16 lanes of each half-wave -> ds_swizzle max-reduce. */
  const int b = g >> 10, s = g & (S_ - 1);
#pragma unroll
  for (int v = 0; v < 8; ++v) {
    int ch = mB + v + 8 * half;
    float sc = sSc2[ch], sh = sSh2[ch], bb = b2[ch];
    float p0 = fmaxf((acc0[v] + bb) * sc + sh, 0.f);
    float p1 = fmaxf((acc1[v] + bb) * sc + sh, 0.f);
    float mx = rmax16(fmaxf(p0, p1));
    if (l == 0) feat[((size_t)b * 128 + ch) * S_ + s] = mx;
  }
}

/* ============================== launcher ============================== */
extern "C" void kernel_launch(void* const* d_in, const int* in_sizes, int n_in,
                              void* d_out, int out_size, void* d_ws, size_t ws_size,
                              hipStream_t stream) {
  (void)in_sizes; (void)n_in; (void)out_size; (void)ws_size;
  const float* xyz = (const float*)d_in[0];
  const float* pts = (const float*)d_in[1];
  const float* w0 = (const float*)d_in[2];
  const float* b0 = (const float*)d_in[3];
  const float* g0 = (const float*)d_in[4];
  const float* bt0 = (const float*)d_in[5];
  const float* w1 = (const float*)d_in[6];
  const float* b1 = (const float*)d_in[7];
  const float* g1 = (const float*)d_in[8];
  const float* bt1 = (const float*)d_in[9];
  const float* w2 = (const float*)d_in[10];
  const float* b2 = (const float*)d_in[11];
  const float* g2 = (const float*)d_in[12];
  const float* bt2 = (const float*)d_in[13];

  float* out = (float*)d_out;                 /* [B,3,S] then [B,128,S] */
  char* wsb = (char*)d_ws;
  int* fpsIdx = (int*)(wsb + OFF_FPS);
  float* stats = (float*)(wsb + OFF_STATS);
  float* X0 = (float*)(wsb + OFF_X0);
  float* Y0 = (float*)(wsb + OFF_Y0);
  float* Y1 = (float*)(wsb + OFF_Y1);
  float* sum0 = stats + 0,   *sqs0 = stats + 64;
  float* sum1 = stats + 128, *sqs1 = stats + 192;
  float* sum2 = stats + 256, *sqs2 = stats + 384;

  zero_stats_kernel<<<1, 512, 0, stream>>>(stats);
  fps_kernel<<<B_, 1024, 0, stream>>>(xyz, fpsIdx, out);
  group_kernel<<<(B_ * S_) / 8, 256, 0, stream>>>(xyz, pts, fpsIdx, X0);
  gemm0_kernel<<<NCOL / 64, 256, 0, stream>>>(X0, w0, b0, Y0, sum0, sqs0);
  gemm1_kernel<<<NCOL / 64, 256, 0, stream>>>(Y0, w1, b1, g0, bt0, sum0, sqs0,
                                              Y1, sum1, sqs1);
  gemm2_stats_kernel<<<NCOL / 64, 256, 0, stream>>>(Y1, w2, b2, g1, bt1, sum1, sqs1,
                                                    sum2, sqs2);
  final_kernel<<<B_ * S_, 256, 0, stream>>>(Y1, w2, b2, g1, bt1, sum1, sqs1,
                                            g2, bt2, sum2, sqs2,
                                            out + (size_t)B_ * 3 * S_);
}